// MarkerGAT_4372276707689
// MI455X (gfx1250) — compile-verified
//
#include <hip/hip_runtime.h>
#include <hip/hip_bf16.h>

// ---- problem constants (match reference) ----
#define N_NODES 50000
#define N_EDGES 800000
#define IN_C    128
#define HID     64
#define HEADS   4
#define OUT_C   64
#define NEG_SLOPE 0.2f

typedef __attribute__((ext_vector_type(2))) float v2f;
typedef __attribute__((ext_vector_type(8))) float v8f;

// ---------------------------------------------------------------------------
// WMMA fp32 GEMM: C[M,Ncols] = A[M,K] * B[K,Ncols], one wave per 16x16 tile.
// Uses V_WMMA_F32_16X16X4_F32 (native fp32 matrix pipe; keeps fp32 fidelity).
// A-frag layout (16x4):  lanes 0-15 -> M=lane, K = k0+{0,1}; lanes 16-31 -> K = k0+{2,3}
// B-frag layout (4x16):  lanes 0-15 -> N=lane, K = k0+{0,1}; lanes 16-31 -> K = k0+{2,3}
// C/D (16x16 f32, 8 VGPRs): vgpr v -> M = v + (lane>=16 ? 8 : 0), N = lane&15
// Grid must be sized so every wave has a full tile (no divergence around WMMA).
// ---------------------------------------------------------------------------
__global__ __launch_bounds__(128) void gat_gemm_wmma_f32(
    const float* __restrict__ A, const float* __restrict__ B,
    float* __restrict__ C, int K, int Ncols, int colTiles) {
  const int wave = blockIdx.x * (blockDim.x >> 5) + (threadIdx.x >> 5);
  const int lane = threadIdx.x & 31;
  const int half = lane >> 4;      // 0: lanes 0-15, 1: lanes 16-31
  const int l16  = lane & 15;
  const int m0 = (wave / colTiles) * 16;
  const int n0 = (wave % colTiles) * 16;

  v8f acc = {};
  const float* arow = A + (size_t)(m0 + l16) * K;
  const float* bcol = B + (n0 + l16);
  for (int k0 = 0; k0 < K; k0 += 4) {
    v2f a, b;
    const int ka = k0 + 2 * half;
    a.x = arow[ka];
    a.y = arow[ka + 1];
    b.x = bcol[(size_t)ka * Ncols];
    b.y = bcol[(size_t)(ka + 1) * Ncols];
    acc = __builtin_amdgcn_wmma_f32_16x16x4_f32(
        /*neg_a=*/false, a, /*neg_b=*/false, b,
        /*c_mod=*/(short)0, acc, /*reuse_a=*/false, /*reuse_b=*/false);
  }
  float* crow = C + (size_t)(m0 + 8 * half) * Ncols + n0 + l16;
#pragma unroll
  for (int v = 0; v < 8; ++v) crow[(size_t)v * Ncols] = acc[v];
}

// ---------------------------------------------------------------------------
// Attention scores: a_src[n,h] = <h[n,h,:], att_src[h,:]>, same for dst.
// One thread per (node, head).
// ---------------------------------------------------------------------------
__global__ void gat_attn_scores(const float* __restrict__ h,
                                const float* __restrict__ att_s,
                                const float* __restrict__ att_d,
                                float* __restrict__ as_, float* __restrict__ ad_,
                                int nNodes, int heads, int hc) {
  int i = blockIdx.x * blockDim.x + threadIdx.x;  // i = n*heads + hd
  if (i >= nNodes * heads) return;
  int hd = i % heads;
  const float* hr = h + (size_t)i * hc;
  const float* s = att_s + (size_t)hd * hc;
  const float* d = att_d + (size_t)hd * hc;
  float accS = 0.f, accD = 0.f;
  for (int c = 0; c < hc; ++c) {
    float v = hr[c];
    accS += v * s[c];
    accD += v * d[c];
  }
  as_[i] = accS;
  ad_[i] = accD;
}

// monotone float -> uint key (order-preserving, handles negatives)
__device__ __forceinline__ unsigned int f2key(float f) {
  unsigned int b = __float_as_uint(f);
  return (b & 0x80000000u) ? ~b : (b | 0x80000000u);
}
__device__ __forceinline__ float key2f(unsigned int k) {
  unsigned int b = (k & 0x80000000u) ? (k & 0x7fffffffu) : ~k;
  return __uint_as_float(b);
}

__device__ __forceinline__ void edge_sd(const long long* __restrict__ ei,
                                        int e, long long& s, long long& d) {
  if (e < N_EDGES) { s = ei[e]; d = ei[N_EDGES + e]; }
  else             { s = e - N_EDGES; d = s; }
}

// Pass A: segment max of leaky_relu(a_src[src]+a_dst[dst]) into m[dst] (uint keys)
__global__ void gat_edge_max(const long long* __restrict__ ei,
                             const float* __restrict__ as_, const float* __restrict__ ad_,
                             unsigned int* __restrict__ m, int tot, int heads) {
  int e = blockIdx.x * blockDim.x + threadIdx.x;
  if (e >= tot) return;
  long long s, d;
  edge_sd(ei, e, s, d);
  for (int hd = 0; hd < heads; ++hd) {
    float t = as_[s * heads + hd] + ad_[d * heads + hd];
    t = t > 0.f ? t : NEG_SLOPE * t;
    atomicMax(&m[d * heads + hd], f2key(t));
  }
}

// Pass B: denom[dst] += exp(e - m[dst])
__global__ void gat_edge_sum(const long long* __restrict__ ei,
                             const float* __restrict__ as_, const float* __restrict__ ad_,
                             const unsigned int* __restrict__ m, float* __restrict__ denom,
                             int tot, int heads) {
  int e = blockIdx.x * blockDim.x + threadIdx.x;
  if (e >= tot) return;
  long long s, d;
  edge_sd(ei, e, s, d);
  for (int hd = 0; hd < heads; ++hd) {
    float t = as_[s * heads + hd] + ad_[d * heads + hd];
    t = t > 0.f ? t : NEG_SLOPE * t;
    float w = __expf(t - key2f(m[d * heads + hd]));
    atomicAdd(&denom[d * heads + hd], w);
  }
}

// Pass C: out[dst, h*HC+c] += alpha[e,h] * h[src, h*HC+c]. One wave32 per edge.
template <int HEADS_, int HC>
__global__ __launch_bounds__(256) void gat_edge_aggregate(
    const long long* __restrict__ ei, const float* __restrict__ h,
    const float* __restrict__ as_, const float* __restrict__ ad_,
    const unsigned int* __restrict__ m, const float* __restrict__ denom,
    float* __restrict__ out, int tot) {
  const int lane = threadIdx.x & 31;
  const int e = blockIdx.x * (blockDim.x >> 5) + (threadIdx.x >> 5);
  if (e >= tot) return;
  long long s, d;
  edge_sd(ei, e, s, d);
  float alpha[HEADS_];
#pragma unroll
  for (int hd = 0; hd < HEADS_; ++hd) {
    float t = as_[s * HEADS_ + hd] + ad_[d * HEADS_ + hd];
    t = t > 0.f ? t : NEG_SLOPE * t;
    float w = __expf(t - key2f(m[d * HEADS_ + hd]));
    alpha[hd] = w / (denom[d * HEADS_ + hd] + 1e-16f);
  }
  const float* hs = h + s * (size_t)(HEADS_ * HC);
  float* od = out + d * (size_t)(HEADS_ * HC);
  for (int c = lane; c < HEADS_ * HC; c += 32)
    atomicAdd(&od[c], alpha[c / HC] * hs[c]);
}

// out[i] = elu(out[i] + bias[i % C])
__global__ void gat_bias_elu(float* __restrict__ x, const float* __restrict__ b,
                             int total, int C) {
  int i = blockIdx.x * blockDim.x + threadIdx.x;
  if (i >= total) return;
  float v = x[i] + b[i % C];
  x[i] = v > 0.f ? v : (__expf(v) - 1.f);
}

extern "C" void kernel_launch(void* const* d_in, const int* in_sizes, int n_in,
                              void* d_out, int out_size, void* d_ws, size_t ws_size,
                              hipStream_t stream) {
  const float* x        = (const float*)d_in[0];
  const long long* ei   = (const long long*)d_in[1];   // int64 [2, E]
  // d_in[2] = edge_attr (ignored by reference: edge_dim=None)
  const float* W1       = (const float*)d_in[3];
  const float* att_src1 = (const float*)d_in[4];
  const float* att_dst1 = (const float*)d_in[5];
  const float* b1       = (const float*)d_in[6];
  const float* W2       = (const float*)d_in[7];
  const float* att_src2 = (const float*)d_in[8];
  const float* att_dst2 = (const float*)d_in[9];
  const float* b2       = (const float*)d_in[10];
  float* out = (float*)d_out;

  const int TOT = N_EDGES + N_NODES;  // self-loops appended

  // ---- workspace layout (bytes) ----
  char* ws = (char*)d_ws;
  size_t off = 0;
  auto alloc = [&](size_t bytes) { char* p = ws + off; off += (bytes + 255) & ~(size_t)255; return p; };
  float*        h1     = (float*)alloc((size_t)N_NODES * HEADS * HID * 4);  // 51.2 MB
  float*        out1   = (float*)alloc((size_t)N_NODES * HEADS * HID * 4);  // 51.2 MB
  float*        h2     = (float*)alloc((size_t)N_NODES * OUT_C * 4);        // 12.8 MB
  float*        aS1    = (float*)alloc((size_t)N_NODES * HEADS * 4);
  float*        aD1    = (float*)alloc((size_t)N_NODES * HEADS * 4);
  unsigned int* m1     = (unsigned int*)alloc((size_t)N_NODES * HEADS * 4);
  float*        den1   = (float*)alloc((size_t)N_NODES * HEADS * 4);
  float*        aS2    = (float*)alloc((size_t)N_NODES * 4);
  float*        aD2    = (float*)alloc((size_t)N_NODES * 4);
  unsigned int* m2     = (unsigned int*)alloc((size_t)N_NODES * 4);
  float*        den2   = (float*)alloc((size_t)N_NODES * 4);

  // =================== layer 1 ===================
  // h1 = x @ W1   (50000x128 * 128x256); 3125 row tiles x 16 col tiles = 50000 waves
  gat_gemm_wmma_f32<<<12500, 128, 0, stream>>>(x, W1, h1, IN_C, HEADS * HID, 16);

  gat_attn_scores<<<(N_NODES * HEADS + 255) / 256, 256, 0, stream>>>(
      h1, att_src1, att_dst1, aS1, aD1, N_NODES, HEADS, HID);

  hipMemsetAsync(m1,   0, (size_t)N_NODES * HEADS * 4, stream);  // key 0 == -inf
  hipMemsetAsync(den1, 0, (size_t)N_NODES * HEADS * 4, stream);
  hipMemsetAsync(out1, 0, (size_t)N_NODES * HEADS * HID * 4, stream);

  gat_edge_max<<<(TOT + 255) / 256, 256, 0, stream>>>(ei, aS1, aD1, m1, TOT, HEADS);
  gat_edge_sum<<<(TOT + 255) / 256, 256, 0, stream>>>(ei, aS1, aD1, m1, den1, TOT, HEADS);
  gat_edge_aggregate<HEADS, HID><<<(TOT + 7) / 8, 256, 0, stream>>>(
      ei, h1, aS1, aD1, m1, den1, out1, TOT);

  gat_bias_elu<<<(N_NODES * HEADS * HID + 255) / 256, 256, 0, stream>>>(
      out1, b1, N_NODES * HEADS * HID, HEADS * HID);

  // =================== layer 2 ===================
  // h2 = out1 @ W2  (50000x256 * 256x64); 3125 row tiles x 4 col tiles = 12500 waves
  gat_gemm_wmma_f32<<<3125, 128, 0, stream>>>(out1, W2, h2, HEADS * HID, OUT_C, 4);

  gat_attn_scores<<<(N_NODES + 255) / 256, 256, 0, stream>>>(
      h2, att_src2, att_dst2, aS2, aD2, N_NODES, 1, OUT_C);

  hipMemsetAsync(m2,   0, (size_t)N_NODES * 4, stream);
  hipMemsetAsync(den2, 0, (size_t)N_NODES * 4, stream);
  hipMemsetAsync(out,  0, (size_t)N_NODES * OUT_C * 4, stream);

  gat_edge_max<<<(TOT + 255) / 256, 256, 0, stream>>>(ei, aS2, aD2, m2, TOT, 1);
  gat_edge_sum<<<(TOT + 255) / 256, 256, 0, stream>>>(ei, aS2, aD2, m2, den2, TOT, 1);
  gat_edge_aggregate<1, OUT_C><<<(TOT + 7) / 8, 256, 0, stream>>>(
      ei, h2, aS2, aD2, m2, den2, out, TOT);

  gat_bias_elu<<<(N_NODES * OUT_C + 255) / 256, 256, 0, stream>>>(
      out, b2, N_NODES * OUT_C, OUT_C);
}